// KGAT_hake_9105330667542
// MI455X (gfx1250) — compile-verified
//
#include <hip/hip_runtime.h>
#include <math.h>

// ---------------------------------------------------------------- constants
#define N_NODES 100000
#define N_EDGES 1000000
#define OUT_COLS 240          // 128 + 64 + 32 + 16
#define LRELU_SLOPE 0.01f

static __device__ __constant__ const float kPI       = 3.1415926235897933f;
static __device__ __constant__ const float kEmbRange = (12.0f + 2.0f) / 64.0f; // (GAMMA+2)/D

typedef __attribute__((ext_vector_type(2))) float v2f;
typedef __attribute__((ext_vector_type(8))) float v8f;

#if defined(__has_builtin)
#  if __has_builtin(__builtin_amdgcn_wmma_f32_16x16x4_f32)
#    define HAVE_WMMA_F32X4 1
#  endif
#endif

// ---------------------------------------------------------------- helpers
__device__ __forceinline__ void atomicMaxFloat(float* addr, float val) {
    // Works for mixed signs: int ordering == float ordering for non-negatives,
    // uint ordering is reversed for negatives.
    if (val >= 0.0f) atomicMax((int*)addr, __float_as_int(val));
    else             atomicMin((unsigned int*)addr, __float_as_uint(val));
}

__device__ __forceinline__ float waveReduceSum(float v) {
    #pragma unroll
    for (int off = 16; off > 0; off >>= 1) v += __shfl_xor(v, off, 32);
    return v;
}

// ---------------------------------------------------------------- kernels
__global__ __launch_bounds__(256) void init_segments(float* amax, float* den) {
    int i = blockIdx.x * blockDim.x + threadIdx.x;
    if (i < N_NODES) { amax[i] = -INFINITY; den[i] = 0.0f; }
}

__global__ __launch_bounds__(256) void zero_f32(float* p, int n) {
    int i = blockIdx.x * blockDim.x + threadIdx.x;
    if (i < n) p[i] = 0.0f;
}

// One wave32 per edge; lane handles dims {lane, lane+32} of D=64.
__global__ __launch_bounds__(256)
void edge_att_raw(const float* __restrict__ ent, const float* __restrict__ rel,
                  const float* __restrict__ phase_w, const float* __restrict__ mod_w,
                  const int* __restrict__ src, const int* __restrict__ dst,
                  const int* __restrict__ etype,
                  float* __restrict__ att, float* __restrict__ amax) {
    int gid  = blockIdx.x * blockDim.x + threadIdx.x;
    int e    = gid >> 5;
    int lane = gid & 31;
    if (e >= N_EDGES) return;

    int hn = dst[e], tn = src[e], rn = etype[e];
    const float* H = ent + (size_t)hn * 128;
    const float* T = ent + (size_t)tn * 128;
    const float* R = rel + (size_t)rn * 192;

    const float inv_scale = kPI / kEmbRange;   // phase = diff / (EMB_RANGE/PI)
    float modsq = 0.0f, psum = 0.0f;
    #pragma unroll
    for (int i = 0; i < 2; ++i) {
        int d = lane + i * 32;
        // modulus part
        float mh = H[64 + d], mt = T[64 + d];
        float mr = fabsf(R[64 + d]);
        float br = fminf(R[128 + d], 1.0f);
        br = (br < -mr) ? -mr : br;
        float md = mh * (mr + br) - mt * (1.0f - br);
        modsq += md * md;
        // phase part
        float phase = (H[d] + R[d] - T[d]) * inv_scale;
        psum += fabsf(__sinf(0.5f * phase));
    }
    modsq = waveReduceSum(modsq);
    psum  = waveReduceSum(psum);
    if (lane == 0) {
        float a = psum * phase_w[0] + sqrtf(modsq) * mod_w[0];
        att[e] = a;
        atomicMaxFloat(&amax[hn], a);
    }
}

__global__ __launch_bounds__(256)
void edge_exp(float* __restrict__ att, const float* __restrict__ amax,
              const int* __restrict__ dst, float* __restrict__ den) {
    int e = blockIdx.x * blockDim.x + threadIdx.x;
    if (e >= N_EDGES) return;
    int d = dst[e];
    float ex = __expf(att[e] - amax[d]);
    att[e] = ex;
    atomicAdd(&den[d], ex);
}

__global__ __launch_bounds__(256)
void edge_softmax_norm(float* __restrict__ att, const float* __restrict__ den,
                       const int* __restrict__ dst) {
    int e = blockIdx.x * blockDim.x + threadIdx.x;
    if (e >= N_EDGES) return;
    att[e] = att[e] / den[dst[e]];
}

// One wave32 per edge: Nh[dst] += node[src] * att.  Rows are 128/64/32 floats,
// contiguous -> coalesced loads; atomics resolve in the 192MB L2.
__global__ __launch_bounds__(256)
void edge_aggregate(const float* __restrict__ node, const float* __restrict__ att,
                    const int* __restrict__ src, const int* __restrict__ dst,
                    float* __restrict__ Nh, int din) {
    int gid  = blockIdx.x * blockDim.x + threadIdx.x;
    int e    = gid >> 5;
    int lane = gid & 31;
    if (e >= N_EDGES) return;
    float a = att[e];
    const float* X = node + (size_t)src[e] * din;
    float*       Y = Nh   + (size_t)dst[e] * din;
    for (int j = lane; j < din; j += 32) atomicAdd(&Y[j], X[j] * a);
}

// One wave32 computes one 16x16 output tile:
//   o = lrelu((node+Nh)@W1^T + b1) + lrelu((node*Nh)@W2^T + b2)
// K-loop in steps of 4 with V_WMMA_F32_16X16X4_F32 (fp32-exact).
// A 16x4 f32 layout: lanes 0-15 hold K=0,1 (VGPR0,1); lanes 16-31 hold K=2,3.
// C/D 16x16 f32 layout: VGPR r -> M=r (lanes 0-15), M=r+8 (lanes 16-31), N=lane&15.
__global__ __launch_bounds__(32)
void gnn_layer_gemm(const float* __restrict__ node, const float* __restrict__ Nh,
                    const float* __restrict__ W1, const float* __restrict__ b1,
                    const float* __restrict__ W2, const float* __restrict__ b2,
                    float* __restrict__ o, int din, int dout) {
    const int lane  = threadIdx.x;            // 0..31, wave32
    const int m0    = blockIdx.x * 16;        // node tile
    const int n0    = blockIdx.y * 16;        // output-feature tile
    const int mrow  = m0 + (lane & 15);
    const int khalf = (lane >> 4) * 2;        // 0 or 2
    const int ncol  = n0 + (lane & 15);

    const float* nodeRow = node + (size_t)mrow * din;
    const float* nhRow   = Nh   + (size_t)mrow * din;
    const float* w1Row   = W1   + (size_t)ncol * din;   // B[k][n] = W[n][k]
    const float* w2Row   = W2   + (size_t)ncol * din;

#if HAVE_WMMA_F32X4
    v8f acc1 = {};
    v8f acc2 = {};
    for (int k = 0; k < din; k += 4) {
        float x0 = nodeRow[k + khalf], x1 = nodeRow[k + khalf + 1];
        float h0 = nhRow[k + khalf],   h1 = nhRow[k + khalf + 1];
        v2f a1; a1.x = x0 + h0; a1.y = x1 + h1;     // (node + Nh) slice
        v2f a2; a2.x = x0 * h0; a2.y = x1 * h1;     // (node * Nh) slice
        v2f bb1; bb1.x = w1Row[k + khalf]; bb1.y = w1Row[k + khalf + 1];
        v2f bb2; bb2.x = w2Row[k + khalf]; bb2.y = w2Row[k + khalf + 1];
        acc1 = __builtin_amdgcn_wmma_f32_16x16x4_f32(false, a1, false, bb1,
                                                     (short)0, acc1, false, false);
        acc2 = __builtin_amdgcn_wmma_f32_16x16x4_f32(false, a2, false, bb2,
                                                     (short)0, acc2, false, false);
    }
    const float bias1 = b1[ncol], bias2 = b2[ncol];
    const int rbase = m0 + ((lane >> 4) << 3);
    #pragma unroll
    for (int r = 0; r < 8; ++r) {
        float c1 = acc1[r] + bias1;
        float c2 = acc2[r] + bias2;
        c1 = (c1 > 0.0f) ? c1 : LRELU_SLOPE * c1;
        c2 = (c2 > 0.0f) ? c2 : LRELU_SLOPE * c2;
        o[(size_t)(rbase + r) * dout + ncol] = c1 + c2;
    }
#else
    // Fallback: plain FMA (same math, no matrix pipe).
    const float bias1 = b1[ncol], bias2 = b2[ncol];
    const int rbase = m0 + ((lane >> 4) << 3);
    for (int r = 0; r < 8; ++r) {
        const float* nr = node + (size_t)(rbase + r) * din;
        const float* hr = Nh   + (size_t)(rbase + r) * din;
        float s1 = 0.0f, s2 = 0.0f;
        for (int k = 0; k < din; ++k) {
            float x = nr[k], h = hr[k];
            s1 = fmaf(x + h, w1Row[k], s1);
            s2 = fmaf(x * h, w2Row[k], s2);
        }
        float c1 = s1 + bias1, c2 = s2 + bias2;
        c1 = (c1 > 0.0f) ? c1 : LRELU_SLOPE * c1;
        c2 = (c2 > 0.0f) ? c2 : LRELU_SLOPE * c2;
        o[(size_t)(rbase + r) * dout + ncol] = c1 + c2;
    }
#endif
}

// One wave32 per node: L2-normalize the row into its output slice.
__global__ __launch_bounds__(256)
void row_normalize(const float* __restrict__ o, float* __restrict__ out,
                   int dout, int colbase) {
    int gid  = blockIdx.x * blockDim.x + threadIdx.x;
    int i    = gid >> 5;
    int lane = gid & 31;
    if (i >= N_NODES) return;
    float ss = 0.0f;
    for (int j = lane; j < dout; j += 32) {
        float v = o[(size_t)i * dout + j];
        ss += v * v;
    }
    ss = waveReduceSum(ss);
    float inv = 1.0f / fmaxf(sqrtf(ss), 1e-12f);
    for (int j = lane; j < dout; j += 32)
        out[(size_t)i * OUT_COLS + colbase + j] = o[(size_t)i * dout + j] * inv;
}

__global__ __launch_bounds__(256)
void copy_ego0(const float* __restrict__ ent, float* __restrict__ out) {
    int idx = blockIdx.x * blockDim.x + threadIdx.x;
    if (idx >= N_NODES * 128) return;
    int i = idx >> 7, j = idx & 127;
    out[(size_t)i * OUT_COLS + j] = ent[idx];
}

// ---------------------------------------------------------------- launch
extern "C" void kernel_launch(void* const* d_in, const int* in_sizes, int n_in,
                              void* d_out, int out_size, void* d_ws, size_t ws_size,
                              hipStream_t stream) {
    (void)in_sizes; (void)n_in; (void)out_size; (void)ws_size;

    const float* ent     = (const float*)d_in[0];
    const float* rel     = (const float*)d_in[1];
    const float* phase_w = (const float*)d_in[2];
    const float* mod_w   = (const float*)d_in[3];
    const float* W1[3]   = {(const float*)d_in[4],  (const float*)d_in[8],  (const float*)d_in[12]};
    const float* B1[3]   = {(const float*)d_in[5],  (const float*)d_in[9],  (const float*)d_in[13]};
    const float* W2[3]   = {(const float*)d_in[6],  (const float*)d_in[10], (const float*)d_in[14]};
    const float* B2[3]   = {(const float*)d_in[7],  (const float*)d_in[11], (const float*)d_in[15]};
    const int*   src     = (const int*)d_in[16];
    const int*   dst     = (const int*)d_in[17];
    const int*   ety     = (const int*)d_in[18];
    float*       out     = (float*)d_out;

    // workspace carve-up (floats): att[E] amax[N] den[N] Nh[N*128] o1[N*64] o2[N*32] o3[N*16]
    float* ws   = (float*)d_ws;
    float* att  = ws;                 ws += N_EDGES;
    float* amax = ws;                 ws += N_NODES;
    float* den  = ws;                 ws += N_NODES;
    float* Nh   = ws;                 ws += (size_t)N_NODES * 128;
    float* oBuf[3];
    oBuf[0] = ws;                     ws += (size_t)N_NODES * 64;
    oBuf[1] = ws;                     ws += (size_t)N_NODES * 32;
    oBuf[2] = ws;                     ws += (size_t)N_NODES * 16;

    const int dims[4] = {128, 64, 32, 16};

    // ---- edge attention + segment softmax
    init_segments<<<(N_NODES + 255) / 256, 256, 0, stream>>>(amax, den);
    edge_att_raw<<<(N_EDGES * 32 + 255) / 256, 256, 0, stream>>>(
        ent, rel, phase_w, mod_w, src, dst, ety, att, amax);
    edge_exp<<<(N_EDGES + 255) / 256, 256, 0, stream>>>(att, amax, dst, den);
    edge_softmax_norm<<<(N_EDGES + 255) / 256, 256, 0, stream>>>(att, den, dst);

    // ---- three GNN layers
    const float* node = ent;
    int colbase = 128;
    for (int l = 0; l < 3; ++l) {
        int din = dims[l], dout = dims[l + 1];
        zero_f32<<<(N_NODES * din + 255) / 256, 256, 0, stream>>>(Nh, N_NODES * din);
        edge_aggregate<<<(N_EDGES * 32 + 255) / 256, 256, 0, stream>>>(
            node, att, src, dst, Nh, din);
        dim3 g(N_NODES / 16, dout / 16);   // 100000 = 6250*16, dims all /16
        gnn_layer_gemm<<<g, 32, 0, stream>>>(node, Nh, W1[l], B1[l], W2[l], B2[l],
                                             oBuf[l], din, dout);
        row_normalize<<<(N_NODES * 32 + 255) / 256, 256, 0, stream>>>(
            oBuf[l], out, dout, colbase);
        colbase += dout;
        node = oBuf[l];
    }

    // ---- out[:, :128] = entity_embed
    copy_ego0<<<(N_NODES * 128 + 255) / 256, 256, 0, stream>>>(ent, out);
}